// MyModel_56968446214682
// MI455X (gfx1250) — compile-verified
//
#include <hip/hip_runtime.h>
#include <hip/hip_bf16.h>

typedef __attribute__((ext_vector_type(2))) float v2f;
typedef __attribute__((ext_vector_type(8))) float v8f;

// CDNA5 has a hardware tanh transcendental (v_tanh_f32). Fall back to a
// 2-trans-op formulation if the builtin is missing.
__device__ __forceinline__ float fast_tanhf(float x) {
#if __has_builtin(__builtin_amdgcn_tanhf)
  return __builtin_amdgcn_tanhf(x);
#else
  float e = __expf(2.0f * x);
  return 1.0f - 2.0f / (e + 1.0f);
#endif
}

// Constants from the reference
#define K_HIGH_PRESSURE 0.0244f
#define I_THRESH        0.39063f
#define KI_CONST        0.1503906f   // interp over constant fp table == const
#define KP_MID          0.3417968f
#define KP_END          0.5f

// Each wave processes 16 rows: lanes n and n+16 both handle row n (duplicate
// loads coalesce). Layer-2 (8x8) is done with two V_WMMA_F32_16X16X4_F32:
//   D[m][n] = sum_k W2[k][m] * h1[row n][k] + b2[m]
// f32 A (16x4) and B (4x16) layouts both split K across lane halves, so the
// per-lane layer-1 outputs land directly in B-operand position with no
// cross-lane data movement.
__global__ __launch_bounds__(256) void pid_mlp_wmma_kernel(
    const float* __restrict__ x,   const float* __restrict__ last,
    const float* __restrict__ kw1, const float* __restrict__ kb1,
    const float* __restrict__ kw2, const float* __restrict__ kb2,
    const float* __restrict__ kw3, const float* __restrict__ kb3,
    float* __restrict__ out, int nrows)
{
  const int lane = threadIdx.x & 31;
  const int wave = threadIdx.x >> 5;
  const int half = lane >> 4;   // which K-half of the WMMA operands this lane holds
  const int m    = lane & 15;   // row-within-tile (B/N index) and A-matrix M index

  // 16 rows per wave, 8 waves per block -> 128 rows per block
  const int row = (blockIdx.x * 8 + wave) * 16 + m;
  const int rc  = row < nrows ? row : (nrows - 1);   // clamp: keep EXEC full

  // ---- streamed inputs (3 of 13 columns + last) -----------------------------
  const size_t xb = (size_t)rc * 13;
  const float hi    = x[xb + 5];
  const float aim   = x[xb + 7];
  const float rate  = x[xb + 11];
  const float lastv = last[rc];

  const float diff = aim - hi;
  const float dhp  = diff * K_HIGH_PRESSURE;

  // ---- layer 1: each lane computes the 4 hidden units of its K-half --------
  // half==0 -> k = {0,1,4,5} ; half==1 -> k = {2,3,6,7}
  const float w1a = half ? kw1[2] : kw1[0];
  const float w1b = half ? kw1[3] : kw1[1];
  const float w1c = half ? kw1[6] : kw1[4];
  const float w1d = half ? kw1[7] : kw1[5];
  const float b1a = half ? kb1[2] : kb1[0];
  const float b1b = half ? kb1[3] : kb1[1];
  const float b1c = half ? kb1[6] : kb1[4];
  const float b1d = half ? kb1[7] : kb1[5];

  const float h0 = fast_tanhf(fmaf(dhp, w1a, b1a));  // B vgpr0 : K = 2*half+0
  const float h1 = fast_tanhf(fmaf(dhp, w1b, b1b));  // B vgpr1 : K = 2*half+1
  const float h2 = fast_tanhf(fmaf(dhp, w1c, b1c));  // B vgpr2 : K = 2*half+4
  const float h3 = fast_tanhf(fmaf(dhp, w1d, b1d));  // B vgpr3 : K = 2*half+5

  // ---- A = W2^T (zero-padded to 16 rows), same K-half striping as B --------
  float a0 = 0.0f, a1 = 0.0f, a2 = 0.0f, a3 = 0.0f;
  if (m < 8) {
    const int k0 = half * 2;
    a0 = kw2[(k0 + 0) * 8 + m];
    a1 = kw2[(k0 + 1) * 8 + m];
    a2 = kw2[(k0 + 4) * 8 + m];
    a3 = kw2[(k0 + 5) * 8 + m];
  }

  // ---- C = b2 broadcast across columns (zero in the padded M=8..15 half) ---
  v8f c;
#pragma unroll
  for (int v = 0; v < 8; ++v) c[v] = half ? 0.0f : kb2[v];

  // ---- two K=4 f32 WMMAs accumulate the full 8x8 layer ----------------------
  v2f A01; A01[0] = a0; A01[1] = a1;
  v2f B01; B01[0] = h0; B01[1] = h1;
  c = __builtin_amdgcn_wmma_f32_16x16x4_f32(false, A01, false, B01,
                                            (short)0, c, false, false);
  v2f A23; A23[0] = a2; A23[1] = a3;
  v2f B23; B23[0] = h2; B23[1] = h3;
  c = __builtin_amdgcn_wmma_f32_16x16x4_f32(false, A23, false, B23,
                                            (short)0, c, false, false);

  // lanes 0-15 now hold h2-preactivation units 0..7 for their row in c[0..7]

  // ---- layer 3: k_out = b3 + sum tanh(c[m]) * w3[m] -------------------------
  float acc = kb3[0];
  acc = fmaf(fast_tanhf(c[0]), kw3[0], acc);
  acc = fmaf(fast_tanhf(c[1]), kw3[1], acc);
  acc = fmaf(fast_tanhf(c[2]), kw3[2], acc);
  acc = fmaf(fast_tanhf(c[3]), kw3[3], acc);
  acc = fmaf(fast_tanhf(c[4]), kw3[4], acc);
  acc = fmaf(fast_tanhf(c[5]), kw3[5], acc);
  acc = fmaf(fast_tanhf(c[6]), kw3[6], acc);
  acc = fmaf(fast_tanhf(c[7]), kw3[7], acc);

  const float dreal = diff * acc;

  // ---- Kp = interp(dhp) * rate. Table is flat in the middle: only the two
  // outer segments [-10,-2.5] and [2.5,10] have non-zero slope. ---------------
  const float u0 = fminf(fmaxf((dhp + 10.0f) * (1.0f / 7.5f), 0.0f), 1.0f);
  const float u4 = fminf(fmaxf((dhp - 2.5f) * (1.0f / 7.5f), 0.0f), 1.0f);
  const float kp = (KP_END + u0 * (KP_MID - KP_END) + u4 * (KP_END - KP_MID)) * rate;
  const float ki = KI_CONST * rate;

  const float ivalue = (fabsf(dhp) < I_THRESH) ? 0.0f : dreal * 0.5f;
  const float result = lastv + ki * ivalue + kp * dreal;

  if (half == 0 && row < nrows) out[row] = result;
}

extern "C" void kernel_launch(void* const* d_in, const int* in_sizes, int n_in,
                              void* d_out, int out_size, void* d_ws, size_t ws_size,
                              hipStream_t stream) {
  const float* x    = (const float*)d_in[0];
  const float* last = (const float*)d_in[1];
  const float* kw1  = (const float*)d_in[2];
  const float* kb1  = (const float*)d_in[3];
  const float* kw2  = (const float*)d_in[4];
  const float* kb2  = (const float*)d_in[5];
  const float* kw3  = (const float*)d_in[6];
  const float* kb3  = (const float*)d_in[7];
  float* out = (float*)d_out;

  const int nrows = in_sizes[0] / 13;
  const int rowsPerBlock = 128;              // 8 waves x 16 rows
  const int grid = (nrows + rowsPerBlock - 1) / rowsPerBlock;
  if (grid > 0) {
    pid_mlp_wmma_kernel<<<grid, 256, 0, stream>>>(
        x, last, kw1, kb1, kw2, kb2, kw3, kb3, out, nrows);
  }
}